// NewLoss_20950850470178
// MI455X (gfx1250) — compile-verified
//
#include <hip/hip_runtime.h>
#include <math.h>

// Problem constants (reference): N=64 batches, C=46 classes, P=2048 frames, MP=256 phones
#define NB 64
#define NC 46
#define NP 2048
#define NMP 256

typedef __attribute__((ext_vector_type(2))) float v2f;
typedef __attribute__((ext_vector_type(8))) float v8f;

// ---------------------------------------------------------------------------
// Kernel 1: phone_out[n,q,c] = sum_p 1[assignment[n,p]==q+1] * input[n,c,p]
// One wave per 16(q) x 16(c) tile, K-loop over p in steps of 4 using the exact
// f32 WMMA (V_WMMA_F32_16X16X4_F32). A is the one-hot indicator built from
// assignment compares; B is streamed from the input rows.
//
// f32 A 16x4 layout (ISA 7.12.2): lanes 0-15 -> M=lane, v0=K0, v1=K1;
//                                 lanes 16-31 -> M=lane-16, v0=K2, v1=K3.
// B 4x16 mirrors it: v0 = row K0 (lanes 0-15) / K2 (lanes 16-31), v1 = K1/K3.
// D 16x16: lane l holds col N=l%16; VGPR r holds row M=r+8*(l/16).
// ---------------------------------------------------------------------------
__global__ __launch_bounds__(256) void seg_sum_wmma(
    const float* __restrict__ input,       // [NB, NC, NP]
    const int*   __restrict__ assignment,  // [NB, NP]
    float*       __restrict__ phone_out)   // [NB, NMP, NC]
{
    const int lane  = threadIdx.x & 31;
    const int wave  = threadIdx.x >> 5;
    const int qt    = blockIdx.x * 8 + wave;   // 0..15 (16 q-tiles of 16)
    const int qBase = qt * 16;
    const int cBase = blockIdx.y * 16;         // 0,16,32 (covers 46 w/ 2 pad)
    const int n     = blockIdx.z;

    const int half  = lane >> 4;               // 0 -> K{0,1}, 1 -> K{2,3}
    const int m     = lane & 15;
    const int qv    = qBase + m + 1;           // assignment value for row M=m
    const int kOff  = half * 2;

    const int c      = cBase + m;
    const int cClamp = (c < NC) ? c : (NC - 1);        // avoid OOB for pad lanes
    const float* __restrict__ brow = input + ((size_t)n * NC + cClamp) * NP;
    const int*   __restrict__ arow = assignment + (size_t)n * NP;

    v8f acc = {};
#pragma unroll 4
    for (int kb = 0; kb < NP; kb += 4) {
        const int p0 = kb + kOff;                       // even -> 8B aligned
        const int2   ap = *(const int2*)  (arow + p0);
        const float2 bp = *(const float2*)(brow + p0);
        v2f a, b;
        a.x = (ap.x == qv) ? 1.0f : 0.0f;
        a.y = (ap.y == qv) ? 1.0f : 0.0f;
        b.x = bp.x;
        b.y = bp.y;
        acc = __builtin_amdgcn_wmma_f32_16x16x4_f32(
            /*neg_a=*/false, a, /*neg_b=*/false, b,
            /*c_mod=*/(short)0, acc, /*reuse_a=*/false, /*reuse_b=*/false);
    }

    if (c < NC) {
#pragma unroll
        for (int r = 0; r < 8; ++r) {
            const int q = qBase + r + half * 8;
            phone_out[((size_t)n * NMP + q) * NC + c] = acc[r];
        }
    }
}

// ---------------------------------------------------------------------------
// Kernel 2: counts[n,q] = #frames with assignment==q+1 (LDS histogram)
// ---------------------------------------------------------------------------
__global__ __launch_bounds__(256) void count_kernel(
    const int* __restrict__ assignment,  // [NB, NP]
    int*       __restrict__ counts)      // [NB, NMP]
{
    __shared__ int cnt[NMP];
    const int n = blockIdx.x;
    const int t = threadIdx.x;
    cnt[t] = 0;
    __syncthreads();
    for (int p = t; p < NP; p += 256) {
        const int a = assignment[(size_t)n * NP + p];
        if (a >= 1 && a <= NMP) atomicAdd(&cnt[a - 1], 1);
    }
    __syncthreads();
    counts[(size_t)n * NMP + t] = cnt[t];
}

// ---------------------------------------------------------------------------
// Kernel 3: per (n,q) row: mean, log-softmax over 46 classes, gather target,
// masked sum. One wave32 per row; shuffle reductions (width = warpSize = 32).
// ---------------------------------------------------------------------------
__global__ __launch_bounds__(256) void finalize_loss(
    const float*         __restrict__ phone_out,  // [NB, NMP, NC]
    const int*           __restrict__ counts,     // [NB, NMP]
    const int*           __restrict__ target,     // [NB, NMP]
    const unsigned char* __restrict__ lab_mask,   // [NB, NMP] (bool)
    float*               __restrict__ out)        // scalar
{
    const int lane = threadIdx.x & 31;
    const int wave = threadIdx.x >> 5;
    const int row  = blockIdx.x * 8 + wave;       // n*NMP + q

    const float inv = 1.0f / fmaxf((float)counts[row], 1.0f);
    const float* __restrict__ pr = phone_out + (size_t)row * NC;

    const float NEG = -3.0e38f;
    const float v0 = (lane      < NC) ? pr[lane]      * inv : NEG;
    const float v1 = (lane + 32 < NC) ? pr[lane + 32] * inv : NEG;

    float mx = fmaxf(v0, v1);
#pragma unroll
    for (int s = 16; s > 0; s >>= 1) mx = fmaxf(mx, __shfl_xor(mx, s, 32));

    float e = ((lane < NC) ? expf(v0 - mx) : 0.0f) +
              ((lane + 32 < NC) ? expf(v1 - mx) : 0.0f);
#pragma unroll
    for (int s = 16; s > 0; s >>= 1) e += __shfl_xor(e, s, 32);

    const float lse = mx + logf(e);

    if (lane == 0) {
        const int   tc   = target[row];
        const float logp = pr[tc] * inv - lse;
        if (lab_mask[row]) atomicAdd(out, -logp);
    }
}

// ---------------------------------------------------------------------------
extern "C" void kernel_launch(void* const* d_in, const int* in_sizes, int n_in,
                              void* d_out, int out_size, void* d_ws, size_t ws_size,
                              hipStream_t stream) {
    const float*         input      = (const float*)d_in[0];          // [64,46,2048]
    const int*           target     = (const int*)d_in[1];            // [64,256]
    const int*           assignment = (const int*)d_in[2];            // [64,2048]
    const unsigned char* lab_mask   = (const unsigned char*)d_in[3];  // [64,256] bool
    float* out = (float*)d_out;

    float* phone_out = (float*)d_ws;                                  // 64*256*46 f32
    int*   counts    = (int*)((char*)d_ws + (size_t)NB * NMP * NC * sizeof(float));

    hipMemsetAsync(d_out, 0, (size_t)out_size * sizeof(float), stream);

    seg_sum_wmma<<<dim3(2, 3, NB), 256, 0, stream>>>(input, assignment, phone_out);
    count_kernel<<<NB, 256, 0, stream>>>(assignment, counts);
    finalize_loss<<<(NB * NMP) / 8, 256, 0, stream>>>(phone_out, counts, target,
                                                      lab_mask, out);
}